// Extrema1D_33938831573315
// MI455X (gfx1250) — compile-verified
//
#include <hip/hip_runtime.h>
#include <stdint.h>

#define W   4096
#define TPB 256
#define EPT (W / TPB)   // 16 elements per thread

typedef unsigned int u32x4 __attribute__((ext_vector_type(4)));
typedef int          i32x8 __attribute__((ext_vector_type(8)));
typedef int          i32x4 __attribute__((ext_vector_type(4)));

#if defined(__HIP_DEVICE_COMPILE__) && \
    __has_builtin(__builtin_amdgcn_tensor_load_to_lds) && \
    __has_builtin(__builtin_amdgcn_tensor_store_from_lds) && \
    __has_builtin(__builtin_amdgcn_s_wait_tensorcnt)
#define USE_TDM 1
#else
#define USE_TDM 0
#endif

__device__ __forceinline__ unsigned long long shfl_xor_u64(unsigned long long v, int m) {
    unsigned lo = __shfl_xor((unsigned)v, m, 32);
    unsigned hi = __shfl_xor((unsigned)(v >> 32), m, 32);
    return ((unsigned long long)hi << 32) | lo;
}

#if USE_TDM
// Build TDM descriptor groups for a 1-row (4096 x f32) tile, per CDNA5 ISA 8.3/8.4.
__device__ __forceinline__ void tdm_groups(unsigned long long gaddr, unsigned lds_addr,
                                           u32x4 &g0, i32x8 &g1) {
    g0[0] = 1u;                                           // count=1, user mode, no gather
    g0[1] = lds_addr;                                     // LDS byte address
    g0[2] = (unsigned)(gaddr & 0xFFFFFFFFull);            // global_addr[31:0]
    g0[3] = (unsigned)((gaddr >> 32) & 0x1FFFFFFull)      // global_addr[56:32]
          | (2u << 30);                                   // type = 2 ("image")
    g1[0] = (int)(2u << 16);                              // workgroup_mask=0, data_size=4B
    g1[1] = (int)((unsigned)W << 16);                     // tensor_dim0[15:0] = 4096
    g1[2] = (int)(1u << 16);                              // tensor_dim0 hi=0, tensor_dim1=1
    g1[3] = (int)((unsigned)W << 16);                     // tile_dim0 = 4096
    g1[4] = 1;                                            // tile_dim1 = 1, tile_dim2 = 0
    g1[5] = W;                                            // tensor_dim0_stride = 4096
    g1[6] = (int)((unsigned)W << 16);                     // tensor_dim1_stride[15:0]
    g1[7] = 0;
}
#endif

__global__ __launch_bounds__(TPB)
void Extrema1D_kernel(const float* __restrict__ x,
                      const int*   __restrict__ dist_p,
                      float*       __restrict__ out) {
    __shared__ float sx[W];                 // input row (TDM load target)
    __shared__ float so[W];                 // output row (TDM store source)
    __shared__ unsigned long long slot[2];  // double-buffered argmax cell

    const int tid  = threadIdx.x;
    const int b    = blockIdx.x;
    const int base = tid * EPT;
    const int dist = dist_p[0];

    const float* xrow = x   + (size_t)b * W;
    float*       orow = out + (size_t)b * W;

    // ---- DMA the row into LDS (wave 0 issues; TDM ignores EXEC) ----
#if USE_TDM
    if (tid < 32) {
        u32x4 g0; i32x8 g1;
        i32x4 gz = {0, 0, 0, 0};
        i32x8 z8 = {0, 0, 0, 0, 0, 0, 0, 0};
        tdm_groups((unsigned long long)(uintptr_t)xrow,
                   (unsigned)(uintptr_t)sx, g0, g1);
        __builtin_amdgcn_tensor_load_to_lds(g0, g1, gz, gz, z8, 0);
        __builtin_amdgcn_s_wait_tensorcnt(0);
    }
#else
    for (int i = tid; i < W / 4; i += TPB)
        ((float4*)sx)[i] = ((const float4*)xrow)[i];
#endif
    __syncthreads();

    // ---- extrema mask + sortable keys, kept in registers ----
    float vx[EPT];
#pragma unroll
    for (int j = 0; j < EPT; ++j) vx[j] = sx[base + j];
    const float xl0  = (base > 0)       ? sx[base - 1]   : 0.0f;
    const float xr16 = (base + EPT < W) ? sx[base + EPT] : 0.0f;

    unsigned mkey[EPT];
#pragma unroll
    for (int j = 0; j < EPT; ++j) {
        const int   i  = base + j;
        const float xc = vx[j];
        const float xl = (j == 0)       ? xl0  : vx[j - 1];
        const float xr = (j == EPT - 1) ? xr16 : vx[j + 1];
        const bool right  = (i < W - 1) && ((xr - xc) > 0.0f);   // pad(dx,[0,1]) > 0
        const bool left   = (i == 0)    || ((xc - xl) <= 0.0f);  // pad(dx,[1,0]) <= 0
        const bool sneg   = (xc <= 0.0f);                        // (1 - sign(x)) as bool
        const bool valley = right && left && sneg;
        const bool peak   = (!right) && (!left) && (!sneg);
        // Monotone map of |x| (>=0): bits | 0x80000000 preserves order, sentinel 0 sorts last.
        mkey[j] = (valley || peak) ? (__float_as_uint(fabsf(xc)) | 0x80000000u) : 0u;
    }

#pragma unroll
    for (int j = 0; j < EPT; ++j) so[base + j] = 0.0f;
    if (tid == 0) { slot[0] = 0ull; slot[1] = 0ull; }
    __syncthreads();

    // ---- greedy NMS: <= ceil(W/(dist+1)) rounds of (argmax, suppress window) ----
    int t = 0;
    for (;;) {
        const int A = t & 1;
        unsigned long long k = 0;
#pragma unroll
        for (int j = 0; j < EPT; ++j) {
            if (mkey[j]) {
                // hi = mapped |x|, lo = ~index  => u64 max == (max |x|, min index)
                const unsigned long long c =
                    ((unsigned long long)mkey[j] << 32) | (unsigned)~(unsigned)(base + j);
                if (c > k) k = c;
            }
        }
#pragma unroll
        for (int m = 1; m < 32; m <<= 1) {
            const unsigned long long o = shfl_xor_u64(k, m);
            if (o > k) k = o;
        }
        if ((tid & 31) == 0 && k) atomicMax(&slot[A], k);   // ds_max_u64, 8 per round
        __syncthreads();
        const unsigned long long best = slot[A];
        __syncthreads();
        if (tid == 0) slot[A] = 0ull;          // safe: next touch of slot[A] is 2 barriers away
        if ((unsigned)(best >> 32) == 0u) break;   // no extrema left

        const int idx = (int)~(unsigned)best;
        if (tid == 0) so[idx] = sx[idx];       // primary keeps its value
#pragma unroll
        for (int j = 0; j < EPT; ++j) {        // kill window in registers
            const int d = base + j - idx;
            if (d >= -dist && d <= dist) mkey[j] = 0u;
        }
        ++t;
    }
    __syncthreads();

    // ---- DMA the result row back out ----
#if USE_TDM
    if (tid < 32) {
        u32x4 g0; i32x8 g1;
        i32x4 gz = {0, 0, 0, 0};
        i32x8 z8 = {0, 0, 0, 0, 0, 0, 0, 0};
        tdm_groups((unsigned long long)(uintptr_t)orow,
                   (unsigned)(uintptr_t)so, g0, g1);
        __builtin_amdgcn_tensor_store_from_lds(g0, g1, gz, gz, z8, 0);
        __builtin_amdgcn_s_wait_tensorcnt(0);
    }
#else
    for (int i = tid; i < W / 4; i += TPB)
        ((float4*)orow)[i] = ((const float4*)so)[i];
#endif
}

extern "C" void kernel_launch(void* const* d_in, const int* in_sizes, int n_in,
                              void* d_out, int out_size, void* d_ws, size_t ws_size,
                              hipStream_t stream) {
    const float* x      = (const float*)d_in[0];
    const int*   dist   = (const int*)d_in[1];
    float*       out    = (float*)d_out;
    const int    rows   = out_size / W;   // B*C = 128
    Extrema1D_kernel<<<rows, TPB, 0, stream>>>(x, dist, out);
}